// QuDe_57277683859570
// MI455X (gfx1250) — compile-verified
//
#include <hip/hip_runtime.h>
#include <hip/hip_bf16.h>

// ---------------------------------------------------------------------------
// U @ x @ U^H  for an 11-qubit hardware-efficient circuit, dim = 2048.
//   ws layout (floats):  Ur | Ui | Yr | Yi   (4 x 2048^2 floats = 64 MB)
//   d_out: complex64 interleaved (2 * 2048^2 floats)
// GEMMs use V_WMMA_F32_16X16X4_F32 with ASYNCcnt-tracked global->LDS DMA
// (global_load_async_to_lds_b128) and double-buffered LDS tiles.
// ---------------------------------------------------------------------------

#define NQ   11
#define DIM  2048
#define PLANE (DIM * DIM)   // 4194304 floats

typedef float v2f __attribute__((ext_vector_type(2)));
typedef float v8f __attribute__((ext_vector_type(8)));

__device__ __forceinline__ v8f wmma_f32(v2f a, v2f b, v8f c) {
    // D = A(16x4,f32) * B(4x16,f32) + C(16x16,f32)
    return __builtin_amdgcn_wmma_f32_16x16x4_f32(
        /*neg_a=*/false, a, /*neg_b=*/false, b,
        /*c_mod=*/(short)0, c, /*reuse_a=*/false, /*reuse_b=*/false);
}

// Async DMA: 16 bytes per lane, global -> LDS, tracked with ASYNCcnt.
__device__ __forceinline__ void async_b128(const float* lds_ptr, const float* gptr) {
    unsigned lds_off = (unsigned)(unsigned long long)lds_ptr;  // low 32 bits = LDS offset
    asm volatile("global_load_async_to_lds_b128 %0, %1, off"
                 :: "v"(lds_off), "v"(gptr) : "memory");
}
__device__ __forceinline__ void wait_async0() {
    asm volatile("s_wait_asynccnt 0" ::: "memory");
}

__device__ __forceinline__ float2 cmul(float2 a, float2 b) {
    return make_float2(a.x * b.x - a.y * b.y, a.x * b.y + a.y * b.x);
}
__device__ __forceinline__ float2 cadd(float2 a, float2 b) {
    return make_float2(a.x + b.x, a.y + b.y);
}

// Composed row permutation of one full CNOT ring (c -> (c+1)%n, c = 0..n-1),
// applied as U = U[p0][p1]...[p10]  =>  old-index = p0(p1(...p10(i))).
__device__ __forceinline__ int ring_perm(int i) {
    #pragma unroll
    for (int c = NQ - 1; c >= 0; --c) {
        int t = (c == NQ - 1) ? 0 : (c + 1);
        i ^= ((i >> (NQ - 1 - c)) & 1) << (NQ - 1 - t);
    }
    return i;
}

// ---------------------------------------------------------------------------
// Kernel 1: build U column-wise entirely in LDS (2 columns / workgroup).
// ---------------------------------------------------------------------------
__global__ __launch_bounds__(256) void qu_build_u(const float* __restrict__ w,
                                                  float* __restrict__ Ur,
                                                  float* __restrict__ Ui) {
    __shared__ float2 col[2][DIM];        // 32 KB
    __shared__ float2 gates[5][NQ][4];    // 55 fused 2x2 gates

    const int tid = threadIdx.x;

    // --- fused gate  gz @ gy @ gx  per (layer, qubit) -----------------------
    if (tid < 5 * NQ) {
        const int L = tid / NQ, q = tid - L * NQ;
        const float s = 0.5f * 0.63245553203367586f;   // 0.5 * sqrt(2/5)
        float hx = w[3 * NQ * L + q] * s;
        float hy = w[3 * NQ * L + NQ + q] * s;
        float hz = w[3 * NQ * L + 2 * NQ + q] * s;
        float cx = __cosf(hx), sx = __sinf(hx);
        float cy = __cosf(hy), sy = __sinf(hy);
        float cz = __cosf(hz), sz = __sinf(hz);
        float2 gx[2][2] = {{{cx, 0.f}, {0.f, -sx}}, {{0.f, -sx}, {cx, 0.f}}};
        float2 gy[2][2] = {{{cy, 0.f}, {-sy, 0.f}}, {{sy, 0.f}, {cy, 0.f}}};
        float2 m1[2][2];
        #pragma unroll
        for (int i = 0; i < 2; ++i)
            #pragma unroll
            for (int j = 0; j < 2; ++j)
                m1[i][j] = cadd(cmul(gy[i][0], gx[0][j]), cmul(gy[i][1], gx[1][j]));
        float2 ez0 = {cz, -sz}, ez1 = {cz, sz};       // e^{-i hz}, e^{+i hz}
        gates[L][q][0] = cmul(ez0, m1[0][0]);
        gates[L][q][1] = cmul(ez0, m1[0][1]);
        gates[L][q][2] = cmul(ez1, m1[1][0]);
        gates[L][q][3] = cmul(ez1, m1[1][1]);
    }
    __syncthreads();

    // --- identity columns ---------------------------------------------------
    const int c0 = blockIdx.x * 2;
    #pragma unroll
    for (int cc = 0; cc < 2; ++cc)
        for (int r = tid; r < DIM; r += 256)
            col[cc][r] = make_float2(r == (c0 + cc) ? 1.f : 0.f, 0.f);
    __syncthreads();

    // --- circuit ------------------------------------------------------------
    for (int L = 0; L < 5; ++L) {
        for (int q = 0; q < NQ; ++q) {
            float2 g00 = gates[L][q][0], g01 = gates[L][q][1];
            float2 g10 = gates[L][q][2], g11 = gates[L][q][3];
            const int bit = 1 << (NQ - 1 - q);
            #pragma unroll
            for (int cc = 0; cc < 2; ++cc)
                for (int p = tid; p < DIM / 2; p += 256) {
                    int r0 = ((p & ~(bit - 1)) << 1) | (p & (bit - 1));
                    int r1 = r0 | bit;
                    float2 a = col[cc][r0], b = col[cc][r1];
                    col[cc][r0] = cadd(cmul(g00, a), cmul(g01, b));
                    col[cc][r1] = cadd(cmul(g10, a), cmul(g11, b));
                }
            __syncthreads();
        }
        if (L < 4) {   // CNOT-ring row permutation, done in-LDS
            #pragma unroll
            for (int cc = 0; cc < 2; ++cc) {
                float2 tmp[8];
                #pragma unroll
                for (int t = 0; t < 8; ++t)
                    tmp[t] = col[cc][ring_perm(tid + t * 256)];
                __syncthreads();
                #pragma unroll
                for (int t = 0; t < 8; ++t)
                    col[cc][tid + t * 256] = tmp[t];
                __syncthreads();
            }
        }
    }

    // --- write planes -------------------------------------------------------
    #pragma unroll
    for (int cc = 0; cc < 2; ++cc) {
        int c = c0 + cc;
        for (int r = tid; r < DIM; r += 256) {
            float2 v = col[cc][r];
            Ur[r * DIM + c] = v.x;
            Ui[r * DIM + c] = v.y;
        }
    }
}

// ---------------------------------------------------------------------------
// GEMM configs: block tile 64(M) x 128(N), K-step 16, 8 waves of 32,
// wave tile 32x32 = 2x2 WMMA(16x16x4) tiles, double-buffered async LDS tiles.
// ---------------------------------------------------------------------------
#define BM 64
#define BN 128
#define BK 16
#define LDA 20     // padded row stride (floats) for M/N-major 16-wide K tiles
#define LDB 132    // padded row stride for the 16x128 NN B tile

// Kernel 2: Yr = Ur @ X, Yi = Ui @ X  (X real, shared B tile)
__global__ __launch_bounds__(256) void qu_gemm_nn_dual(const float* __restrict__ Apr,
                                                       const float* __restrict__ Api,
                                                       const float* __restrict__ B,
                                                       float* __restrict__ Cr,
                                                       float* __restrict__ Ci) {
    __shared__ float Asr[2][BM * LDA];
    __shared__ float Asi[2][BM * LDA];
    __shared__ float Bs[2][BK * LDB];

    const int tid  = threadIdx.x;
    const int lane = tid & 31, wid = tid >> 5;
    const int bm = blockIdx.y * BM, bn = blockIdx.x * BN;
    const int wm = (wid & 1) * 32, wn = (wid >> 1) * 32;

    v8f accR[2][2], accI[2][2];
    const v8f vzero = {0.f, 0.f, 0.f, 0.f, 0.f, 0.f, 0.f, 0.f};
    #pragma unroll
    for (int i = 0; i < 2; ++i)
        #pragma unroll
        for (int j = 0; j < 2; ++j) { accR[i][j] = vzero; accI[i][j] = vzero; }

    const int arow = tid >> 2, acol = (tid & 3) * 4;   // 64x16 A tiles, 1 b128/thread
    const int am = lane & 15, bnl = lane & 15;
    const int b0r = tid >> 5, b0c = (tid & 31) * 4;           // B tile part 0
    const int b1r = (tid + 256) >> 5, b1c = b0c;              // B tile part 1

    auto issue = [&](int buf, int k0) {
        async_b128(&Asr[buf][arow * LDA + acol], &Apr[(bm + arow) * DIM + k0 + acol]);
        async_b128(&Asi[buf][arow * LDA + acol], &Api[(bm + arow) * DIM + k0 + acol]);
        async_b128(&Bs[buf][b0r * LDB + b0c], &B[(k0 + b0r) * DIM + bn + b0c]);
        async_b128(&Bs[buf][b1r * LDB + b1c], &B[(k0 + b1r) * DIM + bn + b1c]);
    };

    issue(0, 0);
    int cur = 0;
    for (int k0 = 0; k0 < DIM; k0 += BK) {
        wait_async0();
        __syncthreads();                       // tile `cur` ready for all waves
        if (k0 + BK < DIM) issue(cur ^ 1, k0 + BK);

        const float* asr = Asr[cur];
        const float* asi = Asi[cur];
        const float* bs  = Bs[cur];
        #pragma unroll
        for (int kk = 0; kk < BK; kk += 4) {
            int ak = kk + ((lane >> 4) << 1);
            v2f ar0 = *(const v2f*)&asr[(wm + am) * LDA + ak];
            v2f ar1 = *(const v2f*)&asr[(wm + 16 + am) * LDA + ak];
            v2f ai0 = *(const v2f*)&asi[(wm + am) * LDA + ak];
            v2f ai1 = *(const v2f*)&asi[(wm + 16 + am) * LDA + ak];
            #pragma unroll
            for (int j = 0; j < 2; ++j) {
                v2f b;
                b.x = bs[ak * LDB + wn + j * 16 + bnl];
                b.y = bs[(ak + 1) * LDB + wn + j * 16 + bnl];
                accR[0][j] = wmma_f32(ar0, b, accR[0][j]);
                accR[1][j] = wmma_f32(ar1, b, accR[1][j]);
                accI[0][j] = wmma_f32(ai0, b, accI[0][j]);
                accI[1][j] = wmma_f32(ai1, b, accI[1][j]);
            }
        }
        __syncthreads();                       // all reads of `cur` done
        cur ^= 1;
    }

    #pragma unroll
    for (int i = 0; i < 2; ++i)
        #pragma unroll
        for (int j = 0; j < 2; ++j)
            #pragma unroll
            for (int v = 0; v < 8; ++v) {
                int row = bm + wm + i * 16 + v + ((lane >> 4) << 3);
                int cn  = bn + wn + j * 16 + (lane & 15);
                Cr[row * DIM + cn] = accR[i][j][v];
                Ci[row * DIM + cn] = accI[i][j][v];
            }
}

// Kernel 3: Z = Y @ conj(U)^T, written interleaved complex to d_out.
//   Zr = Yr*Ur^T + Yi*Ui^T ; Zi = Yi*Ur^T - Yr*Ui^T  (4 WMMA chains -> 2 acc)
__global__ __launch_bounds__(256) void qu_gemm_nt_cplx(const float* __restrict__ Yr,
                                                       const float* __restrict__ Yi,
                                                       const float* __restrict__ Ur,
                                                       const float* __restrict__ Ui,
                                                       float* __restrict__ out) {
    __shared__ float Asr[2][BM * LDA];      // Yr tile 64x16
    __shared__ float Asi[2][BM * LDA];      // Yi tile 64x16
    __shared__ float Bsr[2][BN * LDA];      // Ur tile 128x16 (N-major, K contiguous)
    __shared__ float Bsi[2][BN * LDA];      // Ui tile 128x16

    const int tid  = threadIdx.x;
    const int lane = tid & 31, wid = tid >> 5;
    const int bm = blockIdx.y * BM, bn = blockIdx.x * BN;
    const int wm = (wid & 1) * 32, wn = (wid >> 1) * 32;

    v8f accR[2][2], accI[2][2];
    const v8f vzero = {0.f, 0.f, 0.f, 0.f, 0.f, 0.f, 0.f, 0.f};
    #pragma unroll
    for (int i = 0; i < 2; ++i)
        #pragma unroll
        for (int j = 0; j < 2; ++j) { accR[i][j] = vzero; accI[i][j] = vzero; }

    const int arow = tid >> 2, acol = (tid & 3) * 4;
    const int am = lane & 15, nn = lane & 15;
    const int n0r = tid >> 2, n0c = (tid & 3) * 4;            // B tiles part 0
    const int n1r = (tid + 256) >> 2, n1c = n0c;              // B tiles part 1

    auto issue = [&](int buf, int k0) {
        async_b128(&Asr[buf][arow * LDA + acol], &Yr[(bm + arow) * DIM + k0 + acol]);
        async_b128(&Asi[buf][arow * LDA + acol], &Yi[(bm + arow) * DIM + k0 + acol]);
        async_b128(&Bsr[buf][n0r * LDA + n0c], &Ur[(bn + n0r) * DIM + k0 + n0c]);
        async_b128(&Bsr[buf][n1r * LDA + n1c], &Ur[(bn + n1r) * DIM + k0 + n1c]);
        async_b128(&Bsi[buf][n0r * LDA + n0c], &Ui[(bn + n0r) * DIM + k0 + n0c]);
        async_b128(&Bsi[buf][n1r * LDA + n1c], &Ui[(bn + n1r) * DIM + k0 + n1c]);
    };

    issue(0, 0);
    int cur = 0;
    for (int k0 = 0; k0 < DIM; k0 += BK) {
        wait_async0();
        __syncthreads();
        if (k0 + BK < DIM) issue(cur ^ 1, k0 + BK);

        const float* asr = Asr[cur];
        const float* asi = Asi[cur];
        const float* bsr = Bsr[cur];
        const float* bsi = Bsi[cur];
        #pragma unroll
        for (int kk = 0; kk < BK; kk += 4) {
            int ak = kk + ((lane >> 4) << 1);
            v2f ar0 = *(const v2f*)&asr[(wm + am) * LDA + ak];
            v2f ar1 = *(const v2f*)&asr[(wm + 16 + am) * LDA + ak];
            v2f ai0 = *(const v2f*)&asi[(wm + am) * LDA + ak];
            v2f ai1 = *(const v2f*)&asi[(wm + 16 + am) * LDA + ak];
            v2f arn0 = -ar0, arn1 = -ar1;              // for  -Yr*Ui^T  term
            #pragma unroll
            for (int j = 0; j < 2; ++j) {
                v2f bur = *(const v2f*)&bsr[(wn + j * 16 + nn) * LDA + ak];
                v2f bui = *(const v2f*)&bsi[(wn + j * 16 + nn) * LDA + ak];
                accR[0][j] = wmma_f32(ar0, bur, accR[0][j]);
                accR[0][j] = wmma_f32(ai0, bui, accR[0][j]);
                accR[1][j] = wmma_f32(ar1, bur, accR[1][j]);
                accR[1][j] = wmma_f32(ai1, bui, accR[1][j]);
                accI[0][j] = wmma_f32(ai0, bur, accI[0][j]);
                accI[0][j] = wmma_f32(arn0, bui, accI[0][j]);
                accI[1][j] = wmma_f32(ai1, bur, accI[1][j]);
                accI[1][j] = wmma_f32(arn1, bui, accI[1][j]);
            }
        }
        __syncthreads();
        cur ^= 1;
    }

    #pragma unroll
    for (int i = 0; i < 2; ++i)
        #pragma unroll
        for (int j = 0; j < 2; ++j)
            #pragma unroll
            for (int v = 0; v < 8; ++v) {
                int row = bm + wm + i * 16 + v + ((lane >> 4) << 3);
                int cn  = bn + wn + j * 16 + (lane & 15);
                float2 z = make_float2(accR[i][j][v], accI[i][j][v]);
                *(float2*)&out[(size_t)(row * DIM + cn) * 2] = z;
            }
}

// ---------------------------------------------------------------------------
extern "C" void kernel_launch(void* const* d_in, const int* in_sizes, int n_in,
                              void* d_out, int out_size, void* d_ws, size_t ws_size,
                              hipStream_t stream) {
    const float* x = (const float*)d_in[0];     // 2048 x 2048 f32
    const float* w = (const float*)d_in[1];     // 165 angles
    float* out = (float*)d_out;                 // complex64 interleaved

    float* Ur = (float*)d_ws;
    float* Ui = Ur + PLANE;
    float* Yr = Ui + PLANE;
    float* Yi = Yr + PLANE;

    qu_build_u<<<DIM / 2, 256, 0, stream>>>(w, Ur, Ui);

    dim3 grid(DIM / BN, DIM / BM);              // (16, 32)
    qu_gemm_nn_dual<<<grid, 256, 0, stream>>>(Ur, Ui, x, Yr, Yi);
    qu_gemm_nt_cplx<<<grid, 256, 0, stream>>>(Yr, Yi, Ur, Ui, out);
}